// MoEMultiHead_12927851561601
// MI455X (gfx1250) — compile-verified
//
#include <hip/hip_runtime.h>
#include <hip/hip_bf16.h>

// ---------------------------------------------------------------------------
// MoEMultiHead forward for MI455X (gfx1250, wave32).
// All GEMM-shaped work uses V_WMMA_F32_16X16X4_F32 (native fp32 matrix path,
// keeps reference fp32 precision). Fragment layouts per CDNA5 ISA 7.12.2:
//   A (16x4, MxK): lane l (half=l/16, m=l%16): vgpr0=A[m][2*half], vgpr1=A[m][2*half+1]
//   B (4x16, KxN): lane l (half, n=l%16):      vgpr0=B[2*half][n], vgpr1=B[2*half+1][n]
//   C/D (16x16):   lane l (half, n=l%16):      vgpr r = C[r+8*half][n]
// Attention stages K/V tiles into LDS with GLOBAL_LOAD_ASYNC_TO_LDS_B128
// (double buffered, s_wait_asynccnt) when the toolchain exposes the builtins.
// ---------------------------------------------------------------------------

typedef __attribute__((ext_vector_type(2))) float v2f;
typedef __attribute__((ext_vector_type(8))) float v8f;
typedef __attribute__((ext_vector_type(4))) int v4i;

static __device__ __forceinline__ v8f wmma_f32(v2f a, v2f b, v8f c) {
  return __builtin_amdgcn_wmma_f32_16x16x4_f32(false, a, false, b, (short)0, c,
                                               false, false);
}

#if defined(__HIP_DEVICE_COMPILE__)
#if __has_builtin(__builtin_amdgcn_global_load_async_to_lds_b128) && \
    __has_builtin(__builtin_amdgcn_s_wait_asynccnt)
#define USE_ASYNC_LDS 1
#else
#define USE_ASYNC_LDS 0
#endif
#else
#define USE_ASYNC_LDS 0
#endif

#define B_ 4
#define S_ 1024
#define H_ 256
#define NH_ 4
#define NE_ 4
#define D_ 64

#if USE_ASYNC_LDS
// Copy one 16x64-float (4KB) tile global->LDS: 8 x b128 per lane.
// Builtin signature: (global v4i* src, LDS v4i* dst, imm offset, imm cpol).
static __device__ __forceinline__ void async_copy_tile(const float* __restrict__ g,
                                                       float* l, int lane) {
#pragma unroll
  for (int i = 0; i < 8; ++i) {
    int off = i * 128 + lane * 4;
    __builtin_amdgcn_global_load_async_to_lds_b128(
        (__attribute__((address_space(1))) v4i*)(g + off),
        (__attribute__((address_space(3))) v4i*)(l + off), 0, 0);
  }
}
#endif

// ---------------------------------------------------------------------------
// K1: sx[b,n,s,d] = sum_h x[b,s,h] * hl_w[n,d,h] + hl_b[n,d]
// One wave computes a 16(row) x 64(col) tile. K loop over H=256.
// ---------------------------------------------------------------------------
__global__ void k_proj_sx(const float* __restrict__ x,
                          const float* __restrict__ hl_w,
                          const float* __restrict__ hl_b,
                          float* __restrict__ sx) {
  int bid = blockIdx.x;
  int st = bid % (S_ / 16);
  int t = bid / (S_ / 16);
  int n = t % NH_;
  int b = t / NH_;
  int lane = threadIdx.x;
  int half = lane >> 4, mn = lane & 15;

  const float* xrow = x + ((size_t)b * S_ + (size_t)st * 16) * H_;
  const float* w = hl_w + (size_t)n * D_ * H_;

  v8f acc[4] = {};
  for (int kt = 0; kt < H_; kt += 4) {
    v2f a;
    const float* ap = xrow + (size_t)mn * H_ + kt + 2 * half;
    a.x = ap[0];
    a.y = ap[1];
#pragma unroll
    for (int ct = 0; ct < 4; ++ct) {
      v2f bb;
      const float* wp = w + (size_t)(ct * 16 + mn) * H_ + kt + 2 * half;
      bb.x = wp[0];
      bb.y = wp[1];
      acc[ct] = wmma_f32(a, bb, acc[ct]);
    }
  }
  float* out = sx + (((size_t)b * NH_ + n) * S_ + (size_t)st * 16) * D_;
#pragma unroll
  for (int ct = 0; ct < 4; ++ct)
#pragma unroll
    for (int r = 0; r < 8; ++r) {
      int row = r + 8 * half, col = ct * 16 + mn;
      out[(size_t)row * D_ + col] = acc[ct][r] + hl_b[n * D_ + col];
    }
}

// ---------------------------------------------------------------------------
// K2: q[b,n,e] = sx @ wq[n,e]^T ; k = sx @ wk[n]^T ; v = sx @ wv[n]^T
// sel in [0,6): 0..3 -> expert q, 4 -> k, 5 -> v.  K loop over D=64.
// ---------------------------------------------------------------------------
__global__ void k_qkv(const float* __restrict__ sx, const float* __restrict__ wq,
                      const float* __restrict__ wk, const float* __restrict__ wv,
                      float* __restrict__ q, float* __restrict__ k,
                      float* __restrict__ v) {
  int bid = blockIdx.x;
  int st = bid % (S_ / 16);
  int t = bid / (S_ / 16);
  int sel = t % 6;
  t /= 6;
  int n = t % NH_;
  int b = t / NH_;
  int lane = threadIdx.x;
  int half = lane >> 4, mn = lane & 15;

  const float* a_base = sx + (((size_t)b * NH_ + n) * S_ + (size_t)st * 16) * D_;
  const float* w;
  float* outbase;
  if (sel < NE_) {
    w = wq + (((size_t)n * NE_ + sel) * D_) * D_;
    outbase = q + ((((size_t)b * NH_ + n) * NE_ + sel) * S_ + (size_t)st * 16) * D_;
  } else if (sel == NE_) {
    w = wk + (size_t)n * D_ * D_;
    outbase = k + (((size_t)b * NH_ + n) * S_ + (size_t)st * 16) * D_;
  } else {
    w = wv + (size_t)n * D_ * D_;
    outbase = v + (((size_t)b * NH_ + n) * S_ + (size_t)st * 16) * D_;
  }

  v8f acc[4] = {};
  for (int kt = 0; kt < D_; kt += 4) {
    v2f a;
    const float* ap = a_base + (size_t)mn * D_ + kt + 2 * half;
    a.x = ap[0];
    a.y = ap[1];
#pragma unroll
    for (int ct = 0; ct < 4; ++ct) {
      v2f bb;
      const float* wp = w + (size_t)(ct * 16 + mn) * D_ + kt + 2 * half;
      bb.x = wp[0];
      bb.y = wp[1];
      acc[ct] = wmma_f32(a, bb, acc[ct]);
    }
  }
#pragma unroll
  for (int ct = 0; ct < 4; ++ct)
#pragma unroll
    for (int r = 0; r < 8; ++r) {
      int row = r + 8 * half, col = ct * 16 + mn;
      outbase[(size_t)row * D_ + col] = acc[ct][r];
    }
}

// ---------------------------------------------------------------------------
// K3: flash-style causal attention per (b,n,e); one wave = 16 query rows.
// scores = (Q K^T) * sqrt(D)  (bug-faithful multiply), online softmax,
// P transposed through LDS into A-fragments, P@V via WMMA. Pad mask from sx.
// K/V tiles double-buffered into LDS via async b128 copies when available.
// ---------------------------------------------------------------------------
__global__ void k_attn(const float* __restrict__ q, const float* __restrict__ kk,
                       const float* __restrict__ vv, const float* __restrict__ sx,
                       float* __restrict__ out) {
  __shared__ float plds[16 * 16];
#if USE_ASYNC_LDS
  __shared__ float kvlds[2][2][16 * D_];  // [buf][K/V][4KB]
#endif
  int bid = blockIdx.x;
  int st = bid % (S_ / 16);
  int t = bid / (S_ / 16);
  int e = t % NE_;
  t /= NE_;
  int n = t % NH_;
  int b = t / NH_;
  int lane = threadIdx.x;
  int half = lane >> 4, mn = lane & 15;

  const float* qbase =
      q + ((((size_t)b * NH_ + n) * NE_ + e) * S_ + (size_t)st * 16) * D_;
  const float* kbase = kk + ((size_t)b * NH_ + n) * S_ * D_;
  const float* vbase = vv + ((size_t)b * NH_ + n) * S_ * D_;

  // Preload Q A-fragments for all 16 K-steps (K dim = D = 64).
  v2f qa[16];
#pragma unroll
  for (int kt = 0; kt < 16; ++kt) {
    const float* p = qbase + (size_t)mn * D_ + 4 * kt + 2 * half;
    qa[kt].x = p[0];
    qa[kt].y = p[1];
  }

  float rowmax[8], rowsum[8];
#pragma unroll
  for (int r = 0; r < 8; ++r) {
    rowmax[r] = -__builtin_inff();
    rowsum[r] = 0.f;
  }
  v8f o[4] = {};

  int ntiles = st + 1;  // causal: key tiles 0..st

#if USE_ASYNC_LDS
  async_copy_tile(kbase, kvlds[0][0], lane);
  async_copy_tile(vbase, kvlds[0][1], lane);
#endif

  for (int ktile = 0; ktile < ntiles; ++ktile) {
#if USE_ASYNC_LDS
    if (ktile + 1 < ntiles) {
      // prefetch next tile into the other buffer, then wait for current one
      async_copy_tile(kbase + (size_t)(ktile + 1) * 16 * D_,
                      kvlds[(ktile + 1) & 1][0], lane);
      async_copy_tile(vbase + (size_t)(ktile + 1) * 16 * D_,
                      kvlds[(ktile + 1) & 1][1], lane);
      __builtin_amdgcn_s_wait_asynccnt(16);  // older tile's 16 ops retired
    } else {
      __builtin_amdgcn_s_wait_asynccnt(0);
    }
    __syncthreads();
    const float* ktile_lds = kvlds[ktile & 1][0];
    const float* vtile_lds = kvlds[ktile & 1][1];
#else
    if (ktile + 1 < ntiles)
      __builtin_prefetch(kbase + (size_t)(ktile + 1) * 16 * D_ + lane * 8, 0, 1);
#endif

    // ---- scores tile: 16 WMMAs over K=64 ----
    v8f sc = {};
#pragma unroll
    for (int kt = 0; kt < 16; ++kt) {
      v2f bb;
#if USE_ASYNC_LDS
      const float* p = ktile_lds + (size_t)mn * D_ + 4 * kt + 2 * half;
#else
      const float* p = kbase + (size_t)(ktile * 16 + mn) * D_ + 4 * kt + 2 * half;
#endif
      bb.x = p[0];
      bb.y = p[1];
      sc = wmma_f32(qa[kt], bb, sc);
    }

    bool diag = (ktile == st);
    float p8[8];
#pragma unroll
    for (int r = 0; r < 8; ++r) {
      float sv = sc[r] * 8.0f;  // * sqrt(64), bug-faithful
      if (diag) {
        int qrow = st * 16 + r + 8 * half;
        int kcol = ktile * 16 + mn;
        if (kcol > qrow) sv = -__builtin_inff();
      }
      p8[r] = sv;
    }

    // ---- online softmax update (per-row reduce within 16-lane half) ----
#pragma unroll
    for (int r = 0; r < 8; ++r) {
      float m = p8[r];
#pragma unroll
      for (int ofs = 1; ofs < 16; ofs <<= 1) m = fmaxf(m, __shfl_xor(m, ofs, 32));
      float nm = fmaxf(rowmax[r], m);
      float corr = __expf(rowmax[r] - nm);
      float pe = __expf(p8[r] - nm);
      float ps = pe;
#pragma unroll
      for (int ofs = 1; ofs < 16; ofs <<= 1) ps += __shfl_xor(ps, ofs, 32);
      rowsum[r] = rowsum[r] * corr + ps;
      rowmax[r] = nm;
#pragma unroll
      for (int ct = 0; ct < 4; ++ct) o[ct][r] *= corr;
      p8[r] = pe;
    }

    // ---- transpose P (C layout -> A layout) through LDS ----
    __syncthreads();
#pragma unroll
    for (int r = 0; r < 8; ++r) plds[(r + 8 * half) * 16 + mn] = p8[r];
    __syncthreads();

    // ---- P @ V: K dim = 16 keys -> 4 WMMA K-steps, 4 col tiles ----
#pragma unroll
    for (int kt = 0; kt < 4; ++kt) {
      v2f a;
      a.x = plds[mn * 16 + 4 * kt + 2 * half];
      a.y = plds[mn * 16 + 4 * kt + 2 * half + 1];
#pragma unroll
      for (int ct = 0; ct < 4; ++ct) {
        v2f bb;
#if USE_ASYNC_LDS
        bb.x = vtile_lds[(4 * kt + 2 * half) * D_ + ct * 16 + mn];
        bb.y = vtile_lds[(4 * kt + 2 * half + 1) * D_ + ct * 16 + mn];
#else
        bb.x = vbase[(size_t)(ktile * 16 + 4 * kt + 2 * half) * D_ + ct * 16 + mn];
        bb.y =
            vbase[(size_t)(ktile * 16 + 4 * kt + 2 * half + 1) * D_ + ct * 16 + mn];
#endif
        o[ct] = wmma_f32(a, bb, o[ct]);
      }
    }
  }

  // ---- epilogue: normalize, pad-mask via sx==0, write expert chunk ----
  float* ob = out + (((size_t)b * NH_ + n) * S_ + (size_t)st * 16) * (NE_ * D_) +
              (size_t)e * D_;
  const float* sxb = sx + (((size_t)b * NH_ + n) * S_ + (size_t)st * 16) * D_;
#pragma unroll
  for (int ct = 0; ct < 4; ++ct)
#pragma unroll
    for (int r = 0; r < 8; ++r) {
      int row = r + 8 * half, col = ct * 16 + mn;
      float val = o[ct][r] / rowsum[r];
      if (sxb[(size_t)row * D_ + col] == 0.f) val = 0.f;
      ob[(size_t)row * (NE_ * D_) + col] = val;
    }
}

// ---------------------------------------------------------------------------
// K4: expert gate softmax + mix + bug-faithful residual sx[n,n] + layernorm.
// One wave per (b,n,s) row; 2 d-elements per lane.
// ---------------------------------------------------------------------------
__global__ void k_gate_mix(const float* __restrict__ out,
                           const float* __restrict__ sx,
                           const float* __restrict__ gate_w,
                           const float* __restrict__ gate_b,
                           const float* __restrict__ ln_g,
                           const float* __restrict__ ln_b,
                           float* __restrict__ f_n) {
  int bid = blockIdx.x;
  int s = bid % S_;
  int t = bid / S_;
  int n = t % NH_;
  int b = t / NH_;
  int lane = threadIdx.x;

  const float* row = out + (((size_t)b * NH_ + n) * S_ + s) * (NE_ * D_);

  float gl[NE_];
#pragma unroll
  for (int e2 = 0; e2 < NE_; ++e2) {
    const float* gw = gate_w + ((size_t)n * NE_ + e2) * (NE_ * D_);
    float p = 0.f;
    for (int f = lane; f < NE_ * D_; f += 32) p += row[f] * gw[f];
#pragma unroll
    for (int ofs = 1; ofs < 32; ofs <<= 1) p += __shfl_xor(p, ofs, 32);
    gl[e2] = p + gate_b[n * NE_ + e2];
  }
  float mx = fmaxf(fmaxf(gl[0], gl[1]), fmaxf(gl[2], gl[3]));
  float sm = 0.f;
#pragma unroll
  for (int e2 = 0; e2 < NE_; ++e2) {
    gl[e2] = __expf(gl[e2] - mx);
    sm += gl[e2];
  }
  float inv = 1.f / sm;

  // bug-faithful residual: batch index == head index n
  const float* rs = sx + (((size_t)n * NH_ + n) * S_ + s) * D_;
  float h[2];
#pragma unroll
  for (int i = 0; i < 2; ++i) {
    int d = lane + 32 * i;
    float m = 0.f;
#pragma unroll
    for (int e2 = 0; e2 < NE_; ++e2) m += row[e2 * D_ + d] * gl[e2];
    h[i] = m * inv + rs[d];
  }
  float sum = h[0] + h[1];
#pragma unroll
  for (int ofs = 1; ofs < 32; ofs <<= 1) sum += __shfl_xor(sum, ofs, 32);
  float mean = sum * (1.f / 64.f);
  float vs = (h[0] - mean) * (h[0] - mean) + (h[1] - mean) * (h[1] - mean);
#pragma unroll
  for (int ofs = 1; ofs < 32; ofs <<= 1) vs += __shfl_xor(vs, ofs, 32);
  float rstd = rsqrtf(vs * (1.f / 64.f) + 1e-8f);

  float* o = f_n + (((size_t)b * S_ + s) * NH_ + n) * D_;
#pragma unroll
  for (int i = 0; i < 2; ++i) {
    int d = lane + 32 * i;
    o[d] = (h[i] - mean) * rstd * ln_g[d] + ln_b[d];
  }
}

// ---------------------------------------------------------------------------
// K5: FFN. rows = B*S*NH, feature dim D=64. Two WMMA GEMMs with mask/relu,
// residual, per-row layernorm. Hidden transposed through LDS for GEMM2.
// ---------------------------------------------------------------------------
__global__ void k_ffn(const float* __restrict__ f_n, const float* __restrict__ fc1_w,
                      const float* __restrict__ fc1_b,
                      const float* __restrict__ fc2_w,
                      const float* __restrict__ fc2_b,
                      const float* __restrict__ fnorm_g,
                      const float* __restrict__ fnorm_b, float* __restrict__ fout) {
  __shared__ float hlds[16 * D_];
  int r0 = blockIdx.x * 16;
  int lane = threadIdx.x;
  int half = lane >> 4, mn = lane & 15;
  const float* fx = f_n + (size_t)r0 * D_;

  v8f acc[4] = {};
  for (int kt = 0; kt < D_; kt += 4) {
    v2f a;
    const float* ap = fx + (size_t)mn * D_ + kt + 2 * half;
    a.x = ap[0];
    a.y = ap[1];
#pragma unroll
    for (int ct = 0; ct < 4; ++ct) {
      v2f bb;
      const float* wp = fc1_w + (size_t)(ct * 16 + mn) * D_ + kt + 2 * half;
      bb.x = wp[0];
      bb.y = wp[1];
      acc[ct] = wmma_f32(a, bb, acc[ct]);
    }
  }
#pragma unroll
  for (int ct = 0; ct < 4; ++ct)
#pragma unroll
    for (int r = 0; r < 8; ++r) {
      int row = r + 8 * half, col = ct * 16 + mn;
      float hv = acc[ct][r] + fc1_b[col];
      if (fx[(size_t)row * D_ + col] == 0.f) hv = 0.f;
      hv = fmaxf(hv, 0.f);
      hlds[row * D_ + col] = hv;
    }
  __syncthreads();

  v8f acc2[4] = {};
  for (int kt = 0; kt < D_; kt += 4) {
    v2f a;
    a.x = hlds[mn * D_ + kt + 2 * half];
    a.y = hlds[mn * D_ + kt + 2 * half + 1];
#pragma unroll
    for (int ct = 0; ct < 4; ++ct) {
      v2f bb;
      const float* wp = fc2_w + (size_t)(ct * 16 + mn) * D_ + kt + 2 * half;
      bb.x = wp[0];
      bb.y = wp[1];
      acc2[ct] = wmma_f32(a, bb, acc2[ct]);
    }
  }

  float hv[4][8];
#pragma unroll
  for (int ct = 0; ct < 4; ++ct)
#pragma unroll
    for (int r = 0; r < 8; ++r) {
      int row = r + 8 * half, col = ct * 16 + mn;
      float x0 = fx[(size_t)row * D_ + col];
      float tv = acc2[ct][r] + fc2_b[col];
      if (x0 == 0.f) tv = 0.f;
      hv[ct][r] = tv + x0;
    }

#pragma unroll
  for (int r = 0; r < 8; ++r) {
    float sum = hv[0][r] + hv[1][r] + hv[2][r] + hv[3][r];
#pragma unroll
    for (int ofs = 1; ofs < 16; ofs <<= 1) sum += __shfl_xor(sum, ofs, 32);
    float mean = sum * (1.f / 64.f);
    float vs = 0.f;
#pragma unroll
    for (int ct = 0; ct < 4; ++ct) {
      float dlt = hv[ct][r] - mean;
      vs += dlt * dlt;
    }
#pragma unroll
    for (int ofs = 1; ofs < 16; ofs <<= 1) vs += __shfl_xor(vs, ofs, 32);
    float rstd = rsqrtf(vs * (1.f / 64.f) + 1e-8f);
#pragma unroll
    for (int ct = 0; ct < 4; ++ct) {
      int row = r + 8 * half, col = ct * 16 + mn;
      fout[(size_t)(r0 + row) * D_ + col] =
          (hv[ct][r] - mean) * rstd * fnorm_g[col] + fnorm_b[col];
    }
  }
}

// ---------------------------------------------------------------------------
// K6: head-gate softmax over NH, scale per 64-wide head chunk. One wave/row.
// ---------------------------------------------------------------------------
__global__ void k_fgate(const float* __restrict__ fout,
                        const float* __restrict__ fgate_w, float* __restrict__ y) {
  int row = blockIdx.x;  // b*S + s
  int lane = threadIdx.x;
  const float* fr = fout + (size_t)row * H_;
  float gl[NH_];
#pragma unroll
  for (int j = 0; j < NH_; ++j) {
    const float* gw = fgate_w + (size_t)j * H_;
    float p = 0.f;
    for (int f = lane; f < H_; f += 32) p += fr[f] * gw[f];
#pragma unroll
    for (int ofs = 1; ofs < 32; ofs <<= 1) p += __shfl_xor(p, ofs, 32);
    gl[j] = p;
  }
  float mx = fmaxf(fmaxf(gl[0], gl[1]), fmaxf(gl[2], gl[3]));
  float sm = 0.f;
#pragma unroll
  for (int j = 0; j < NH_; ++j) {
    gl[j] = __expf(gl[j] - mx);
    sm += gl[j];
  }
  float inv = 1.f / sm;
  for (int f = lane; f < H_; f += 32)
    y[(size_t)row * H_ + f] = fr[f] * gl[f / D_] * inv;
}

// ---------------------------------------------------------------------------
extern "C" void kernel_launch(void* const* d_in, const int* in_sizes, int n_in,
                              void* d_out, int out_size, void* d_ws, size_t ws_size,
                              hipStream_t stream) {
  const float* x = (const float*)d_in[0];
  const float* hl_w = (const float*)d_in[1];
  const float* hl_b = (const float*)d_in[2];
  const float* wq = (const float*)d_in[3];
  const float* wk = (const float*)d_in[4];
  const float* wv = (const float*)d_in[5];
  const float* gate_w = (const float*)d_in[6];
  const float* gate_b = (const float*)d_in[7];
  const float* ln_g = (const float*)d_in[8];
  const float* ln_b = (const float*)d_in[9];
  const float* fc1_w = (const float*)d_in[10];
  const float* fc1_b = (const float*)d_in[11];
  const float* fc2_w = (const float*)d_in[12];
  const float* fc2_b = (const float*)d_in[13];
  const float* fnorm_g = (const float*)d_in[14];
  const float* fnorm_b = (const float*)d_in[15];
  const float* fgate_w = (const float*)d_in[16];

  // workspace layout (floats): ~52 MB total
  float* ws = (float*)d_ws;
  float* sx = ws;                                       // B*NH*S*D      = 1M
  float* kbuf = sx + (size_t)B_ * NH_ * S_ * D_;        // 1M
  float* vbuf = kbuf + (size_t)B_ * NH_ * S_ * D_;      // 1M
  float* qbuf = vbuf + (size_t)B_ * NH_ * S_ * D_;      // B*NH*NE*S*D   = 4M
  float* attout = qbuf + (size_t)B_ * NH_ * NE_ * S_ * D_;  // 4M
  float* f_n = attout + (size_t)B_ * NH_ * S_ * NE_ * D_;   // B*S*H = 1M
  float* foutbuf = f_n + (size_t)B_ * S_ * H_;              // 1M

  k_proj_sx<<<B_ * NH_ * (S_ / 16), 32, 0, stream>>>(x, hl_w, hl_b, sx);
  k_qkv<<<B_ * NH_ * 6 * (S_ / 16), 32, 0, stream>>>(sx, wq, wk, wv, qbuf, kbuf,
                                                     vbuf);
  k_attn<<<B_ * NH_ * NE_ * (S_ / 16), 32, 0, stream>>>(qbuf, kbuf, vbuf, sx,
                                                        attout);
  k_gate_mix<<<B_ * NH_ * S_, 32, 0, stream>>>(attout, sx, gate_w, gate_b, ln_g,
                                               ln_b, f_n);
  k_ffn<<<(B_ * S_ * NH_) / 16, 32, 0, stream>>>(f_n, fc1_w, fc1_b, fc2_w, fc2_b,
                                                 fnorm_g, fnorm_b, foutbuf);
  k_fgate<<<B_ * S_, 32, 0, stream>>>(foutbuf, fgate_w, (float*)d_out);
}